// GCGRUCell_44839458570785
// MI455X (gfx1250) — compile-verified
//
#include <hip/hip_runtime.h>
#include <hip/hip_bf16.h>
#include <math.h>

// Problem constants (match reference)
#define N_NODES 6000
#define E_EDGES 192000
#define CIN     32
#define CHID    64
#define DIMS    3
#define KS      5
#define KKER    125          // 5^3
#define SPROD   8            // 2^3
#define KF_X    (KKER*CIN)   // 4000  flattened (k,cin) for x-side convs
#define KF_H    (KKER*CHID)  // 8000  flattened (k,cin) for h-side convs

typedef __attribute__((ext_vector_type(2))) float v2f;
typedef __attribute__((ext_vector_type(8))) float v8f;

// ---------------------------------------------------------------------------
// Kernel 1: edge scatter.  One wave per edge; lanes = input channels.
// Builds T_x [N,125,32], T_h [N,125,64] and float degree counts, all with
// hardware f32 atomics (global_atomic_add_f32).
// ---------------------------------------------------------------------------
__global__ void spline_scatter_kernel(const float* __restrict__ x,
                                      const float* __restrict__ hidden,
                                      const int*   __restrict__ ei,
                                      const float* __restrict__ pseudo,
                                      float* __restrict__ Tx,
                                      float* __restrict__ Th,
                                      float* __restrict__ deg) {
  const int w    = (blockIdx.x * blockDim.x + threadIdx.x) >> 5;   // edge id
  const int lane = threadIdx.x & 31;
  if (w >= E_EDGES) return;

  const int src = ei[w];
  const int dst = ei[E_EDGES + w];

  // open B-spline, degree 1: KS-1 intervals over [0,1]
  float fr[DIMS]; int i0[DIMS];
#pragma unroll
  for (int d = 0; d < DIMS; ++d) {
    float u = pseudo[(size_t)w * DIMS + d] * (float)(KS - 1);
    int   i = (int)floorf(u);
    i = i < 0 ? 0 : (i > KS - 2 ? KS - 2 : i);
    i0[d] = i;
    fr[d] = u - (float)i;
  }

  int   idx[SPROD];
  float bas[SPROD];
#pragma unroll
  for (int s = 0; s < SPROD; ++s) {
    int id = 0; float b = 1.0f;
#pragma unroll
    for (int d = 0; d < DIMS; ++d) {
      const int bit    = (s >> d) & 1;
      const int stride = (d == 0) ? 25 : ((d == 1) ? 5 : 1); // KS^(DIMS-1-d)
      id += (i0[d] + bit) * stride;
      b  *= bit ? fr[d] : (1.0f - fr[d]);
    }
    idx[s] = id; bas[s] = b;
  }

  const float xv  = x[(size_t)src * CIN + lane];
  const float hv0 = hidden[(size_t)src * CHID + lane];
  const float hv1 = hidden[(size_t)src * CHID + 32 + lane];

  float* txd = Tx + (size_t)dst * KF_X;
  float* thd = Th + (size_t)dst * KF_H;
#pragma unroll
  for (int s = 0; s < SPROD; ++s) {
    unsafeAtomicAdd(txd + idx[s] * CIN  + lane,      bas[s] * xv);
    unsafeAtomicAdd(thd + idx[s] * CHID + lane,      bas[s] * hv0);
    unsafeAtomicAdd(thd + idx[s] * CHID + 32 + lane, bas[s] * hv1);
  }
  if (lane == 0) unsafeAtomicAdd(deg + dst, 1.0f);
}

// ---------------------------------------------------------------------------
// WMMA f32 16x16x4 tile contraction.
// A layout (ISA 7.12.2, 32-bit A 16x4): lanes 0-15 = rows, VGPR0/1 = K0/K1;
// lanes 16-31 = rows, VGPR0/1 = K2/K3  -> one contiguous float2 per lane.
// B layout (4x16, row striped across lanes): VGPR0 = {K0 lanes0-15, K2 lanes16-31},
// VGPR1 = {K1, K3}.
// C/D: VGPR j -> row j (lanes 0-15) / row j+8 (lanes 16-31), col = lane&15.
// Computes, for NMAT weight matrices sharing one A stream:
//   O[m] = (T @ W[m]) / max(deg,1) + Xin @ R[m] + B[m]
// for a 16-row x 16-col output tile.
// ---------------------------------------------------------------------------
template <int NMAT>
__device__ __forceinline__ void gemm_tile(const float* __restrict__ T, int Kflat,
                                          const float* __restrict__ Xin, int Cin,
                                          const float* const* Wm,
                                          const float* const* Rm,
                                          const float* const* Bm,
                                          float* const* Om,
                                          const float* __restrict__ deg,
                                          int r0, int cb, int lane) {
  const int row16 = lane & 15;
  const int kh    = lane >> 4;          // 0: K0/K1 half, 1: K2/K3 half
  const int col   = cb + row16;

  v8f acc[NMAT] = {};

  const float* arow = T + (size_t)(r0 + row16) * Kflat + 2 * kh;
  for (int kk = 0; kk < Kflat; kk += 4) {
    v2f a = *(const v2f*)(arow + kk);
    __builtin_prefetch(arow + kk + 128, 0, 0);   // stream next 512B of the T row
#pragma unroll
    for (int m = 0; m < NMAT; ++m) {
      const float* wp = Wm[m] + (size_t)(kk + 2 * kh) * CHID + col;
      v2f b; b.x = wp[0]; b.y = wp[CHID];
      acc[m] = __builtin_amdgcn_wmma_f32_16x16x4_f32(
          false, a, false, b, (short)0, acc[m], false, false);
    }
  }

  // mean aggregation: divide spline part by max(deg,1) (per output row)
  float inv[8];
#pragma unroll
  for (int j = 0; j < 8; ++j) {
    float dg = deg[r0 + j + 8 * kh];
    inv[j] = 1.0f / fmaxf(dg, 1.0f);
  }
#pragma unroll
  for (int m = 0; m < NMAT; ++m)
#pragma unroll
    for (int j = 0; j < 8; ++j) acc[m][j] *= inv[j];

  // root-weight contribution: extra K-chunks of the same WMMA pipeline
  const float* xrow = Xin + (size_t)(r0 + row16) * Cin + 2 * kh;
  for (int kk = 0; kk < Cin; kk += 4) {
    v2f a = *(const v2f*)(xrow + kk);
#pragma unroll
    for (int m = 0; m < NMAT; ++m) {
      const float* rp = Rm[m] + (size_t)(kk + 2 * kh) * CHID + col;
      v2f b; b.x = rp[0]; b.y = rp[CHID];
      acc[m] = __builtin_amdgcn_wmma_f32_16x16x4_f32(
          false, a, false, b, (short)0, acc[m], false, false);
    }
  }

  // bias + store
#pragma unroll
  for (int m = 0; m < NMAT; ++m) {
    const float bs = Bm[m][col];
#pragma unroll
    for (int j = 0; j < 8; ++j)
      Om[m][(size_t)(r0 + j + 8 * kh) * CHID + col] = acc[m][j] + bs;
  }
}

// 3000 waves total: 1500 x-side (T_x against W_xr/W_xz/W_xn), 1500 h-side.
__global__ void spline_gemm_kernel(
    const float* __restrict__ Tx, const float* __restrict__ Th,
    const float* __restrict__ x,  const float* __restrict__ hidden,
    const float* __restrict__ deg,
    const float* Wxr, const float* Rxr, const float* bxr,
    const float* Wxz, const float* Rxz, const float* bxz,
    const float* Wxn, const float* Rxn, const float* bxn,
    const float* Whr, const float* Rhr, const float* bhr,
    const float* Whz, const float* Rhz, const float* bhz,
    float* oxr, float* oxz, float* oxn, float* ohr, float* ohz) {
  const int w    = (blockIdx.x * blockDim.x + threadIdx.x) >> 5;
  const int lane = threadIdx.x & 31;
  const int NT   = N_NODES / 16;              // 375 node tiles

  if (w < NT * 4) {
    const int tile = w >> 2;
    const int cb   = (w & 3) * 16;
    const float* Wm[3] = {Wxr, Wxz, Wxn};
    const float* Rm[3] = {Rxr, Rxz, Rxn};
    const float* Bm[3] = {bxr, bxz, bxn};
    float*       Om[3] = {oxr, oxz, oxn};
    gemm_tile<3>(Tx, KF_X, x, CIN, Wm, Rm, Bm, Om, deg, tile * 16, cb, lane);
  } else {
    const int w2   = w - NT * 4;
    const int tile = w2 >> 2;
    const int cb   = (w2 & 3) * 16;
    const float* Wm[2] = {Whr, Whz};
    const float* Rm[2] = {Rhr, Rhz};
    const float* Bm[2] = {bhr, bhz};
    float*       Om[2] = {ohr, ohz};
    gemm_tile<2>(Th, KF_H, hidden, CHID, Wm, Rm, Bm, Om, deg, tile * 16, cb, lane);
  }
}

// ---------------------------------------------------------------------------
// Kernel 3: GRU gate combine (elementwise over N x CHID).
// r = sig(xr+hr); z = sig(xz+hz); n = tanh(xn + r*hr); h' = (1-z)*n + z*h
// (reference reuses the hr conv output inside n; replicated here)
// ---------------------------------------------------------------------------
__global__ void gru_combine_kernel(const float* __restrict__ oxr,
                                   const float* __restrict__ oxz,
                                   const float* __restrict__ oxn,
                                   const float* __restrict__ ohr,
                                   const float* __restrict__ ohz,
                                   const float* __restrict__ hidden,
                                   float* __restrict__ out) {
  const int t = blockIdx.x * blockDim.x + threadIdx.x;
  if (t >= N_NODES * CHID) return;
  const float hr = ohr[t];
  const float r  = 1.0f / (1.0f + expf(-(oxr[t] + hr)));
  const float z  = 1.0f / (1.0f + expf(-(oxz[t] + ohz[t])));
  const float nn = tanhf(oxn[t] + r * hr);
  out[t] = (1.0f - z) * nn + z * hidden[t];
}

// ---------------------------------------------------------------------------
extern "C" void kernel_launch(void* const* d_in, const int* in_sizes, int n_in,
                              void* d_out, int out_size, void* d_ws, size_t ws_size,
                              hipStream_t stream) {
  const float* x      = (const float*)d_in[0];
  const float* hidden = (const float*)d_in[1];
  const int*   ei     = (const int*)d_in[2];
  const float* ea     = (const float*)d_in[3];
  const float* Wxr = (const float*)d_in[4],  *Rxr = (const float*)d_in[5],  *bxr = (const float*)d_in[6];
  const float* Whr = (const float*)d_in[7],  *Rhr = (const float*)d_in[8],  *bhr = (const float*)d_in[9];
  const float* Wxz = (const float*)d_in[10], *Rxz = (const float*)d_in[11], *bxz = (const float*)d_in[12];
  const float* Whz = (const float*)d_in[13], *Rhz = (const float*)d_in[14], *bhz = (const float*)d_in[15];
  const float* Wxn = (const float*)d_in[16], *Rxn = (const float*)d_in[17], *bxn = (const float*)d_in[18];
  // d_in[19..21] (W_hn/root_hn/b_hn) are unused by the reference forward.

  // workspace layout (floats)
  float* ws  = (float*)d_ws;
  float* Tx  = ws;                                    // 24,000,000
  float* Th  = Tx + (size_t)N_NODES * KF_X;           // 48,000,000
  float* deg = Th + (size_t)N_NODES * KF_H;           // 6,000 (padded to 6400)
  float* oxr = deg + 6400;
  float* oxz = oxr + (size_t)N_NODES * CHID;
  float* oxn = oxz + (size_t)N_NODES * CHID;
  float* ohr = oxn + (size_t)N_NODES * CHID;
  float* ohz = ohr + (size_t)N_NODES * CHID;          // total ~295.8 MB

  const size_t zero_bytes =
      ((size_t)N_NODES * KF_X + (size_t)N_NODES * KF_H + 6400) * sizeof(float);
  hipMemsetAsync(d_ws, 0, zero_bytes, stream);

  // 1 wave per edge, 8 waves per block
  spline_scatter_kernel<<<E_EDGES / 8, 256, 0, stream>>>(x, hidden, ei, ea, Tx, Th, deg);

  // 3000 waves = 375 blocks * 8 waves
  spline_gemm_kernel<<<375, 256, 0, stream>>>(
      Tx, Th, x, hidden, deg,
      Wxr, Rxr, bxr, Wxz, Rxz, bxz, Wxn, Rxn, bxn,
      Whr, Rhr, bhr, Whz, Rhz, bhz,
      oxr, oxz, oxn, ohr, ohz);

  gru_combine_kernel<<<(N_NODES * CHID + 255) / 256, 256, 0, stream>>>(
      oxr, oxz, oxn, ohr, ohz, hidden, (float*)d_out);
}